// MCM_70944269795647
// MI455X (gfx1250) — compile-verified
//
#include <hip/hip_runtime.h>
#include <cstdint>
#include <math.h>

// Problem constants (from reference): x [256,2048] f32, M [2048,2048] f32,
// out[b,i] = max_j M[i,j]*x[b,j].
#define N_DIM   2048
#define B_DIM   256
#define I_TILE  64
#define B_TILE  64
#define KB      64                  // j-chunk staged per double-buffer step
#define K_SPLIT 4                   // split the j-reduction across grid.z
#define K_RANGE (N_DIM / K_SPLIT)   // 512 j per workgroup
#define N_CHUNK (K_RANGE / KB)      // 8 chunks
#define TILE_ELEMS (I_TILE * KB)    // 4096 floats = 16 KB per tile

// Low 32 bits of a generic pointer to a __shared__ object = LDS byte offset
// (flat->LDS mapping truncates to addr[31:0]); this is the VDST operand the
// async-to-LDS instruction wants.
__device__ __forceinline__ uint32_t lds_off(const void* p) {
  return (uint32_t)(uintptr_t)p;
}

// CDNA5 async copy: global -> LDS, 16 bytes per lane, tracked by ASYNCcnt.
__device__ __forceinline__ void async_copy_b128(uint32_t lds_byte, const float* g) {
  asm volatile("global_load_async_to_lds_b128 %0, %1, off"
               :
               : "v"(lds_byte), "v"((uint64_t)(uintptr_t)g)
               : "memory");
}

__device__ __forceinline__ void wait_async_zero() {
  asm volatile("s_wait_asynccnt 0" ::: "memory");
}

// Stage one 64(row) x 64(j) f32 tile of M and of x into LDS.
// Lane map: j4 = tid&15 (16-byte column group), rows tid>>4 + 16*r.
// Each row's 16 lanes read 256 contiguous bytes of global memory (coalesced).
// LDS column is XOR-swizzled (col' = j4 ^ (row>>2)) so that the compute-side
// strided-row ds_load_b128 reads hit 16 distinct bank groups.
__device__ __forceinline__ void stage_tiles(float* ldsM, float* ldsX,
                                            const float* __restrict__ M,
                                            const float* __restrict__ X,
                                            int iBase, int bBase, int k0, int tid) {
  const int j4   = tid & 15;
  const int row0 = tid >> 4;
#pragma unroll
  for (int r = 0; r < 4; ++r) {
    const int row  = row0 + (r << 4);
    const int colp = ((j4 ^ (row >> 2)) << 2);   // swizzled col, in dwords
    async_copy_b128(lds_off(ldsM + row * KB + colp),
                    M + (size_t)(iBase + row) * N_DIM + (size_t)(k0 + (j4 << 2)));
    async_copy_b128(lds_off(ldsX + row * KB + colp),
                    X + (size_t)(bBase + row) * N_DIM + (size_t)(k0 + (j4 << 2)));
  }
}

__global__ __launch_bounds__(256) void tropical_mv(const float* __restrict__ x,
                                                   const float* __restrict__ M,
                                                   float* __restrict__ out) {
  __shared__ __align__(16) float lds[2][2][TILE_ELEMS];  // [buf][M/X][tile] = 64 KB

  const int tid   = threadIdx.x;
  const int iBase = blockIdx.x * I_TILE;
  const int bBase = blockIdx.y * B_TILE;
  const int kBase = blockIdx.z * K_RANGE;

  const int itid = tid & 15;   // i sub-tile: i0 = 4*itid
  const int btid = tid >> 4;   // b sub-tile: b0 = 4*btid

  float acc[4][4];             // acc[b][i]
#pragma unroll
  for (int a = 0; a < 4; ++a)
#pragma unroll
    for (int b = 0; b < 4; ++b) acc[a][b] = -INFINITY;

  // Prologue: stage chunk 0 into buffer 0.
  stage_tiles(lds[0][0], lds[0][1], M, x, iBase, bBase, kBase, tid);

  for (int c = 0; c < N_CHUNK; ++c) {
    wait_async_zero();   // my async writes for chunk c are in LDS
    __syncthreads();     // everyone's are (each wave waits before signaling)

    // Kick off next chunk into the other buffer; overlaps with compute below.
    if (c + 1 < N_CHUNK) {
      stage_tiles(lds[(c + 1) & 1][0], lds[(c + 1) & 1][1], M, x,
                  iBase, bBase, kBase + (c + 1) * KB, tid);
    }

    const float* ldsM = lds[c & 1][0];
    const float* ldsX = lds[c & 1][1];

#pragma unroll
    for (int j4 = 0; j4 < KB / 4; ++j4) {        // 4 j-values per iteration
      const int mcol = ((j4 ^ itid) << 2);       // undo XOR swizzle
      const int xcol = ((j4 ^ btid) << 2);
      float4 mv[4], xv[4];
#pragma unroll
      for (int k = 0; k < 4; ++k) {
        mv[k] = *(const float4*)(ldsM + ((itid << 2) + k) * KB + mcol);
        xv[k] = *(const float4*)(ldsX + ((btid << 2) + k) * KB + xcol);
      }
      // 64 mul + 64 max per iteration: dual-issue VOPD fodder on wave32.
#pragma unroll
      for (int kb = 0; kb < 4; ++kb)
#pragma unroll
        for (int ki = 0; ki < 4; ++ki) {
          acc[kb][ki] = fmaxf(acc[kb][ki], mv[ki].x * xv[kb].x);
          acc[kb][ki] = fmaxf(acc[kb][ki], mv[ki].y * xv[kb].y);
          acc[kb][ki] = fmaxf(acc[kb][ki], mv[ki].z * xv[kb].z);
          acc[kb][ki] = fmaxf(acc[kb][ki], mv[ki].w * xv[kb].w);
        }
    }
    __syncthreads();     // all waves done reading this buffer before reuse
  }

  // Combine the K_SPLIT partial maxima. f32 max is exact & order-independent,
  // so atomics keep the result deterministic.
  const int i0 = iBase + (itid << 2);
  const int b0 = bBase + (btid << 2);
#pragma unroll
  for (int kb = 0; kb < 4; ++kb)
#pragma unroll
    for (int ki = 0; ki < 4; ++ki) {
      float* p = out + (size_t)(b0 + kb) * N_DIM + (size_t)(i0 + ki);
      __hip_atomic_fetch_max(p, acc[kb][ki], __ATOMIC_RELAXED,
                             __HIP_MEMORY_SCOPE_AGENT);
    }
}

__global__ void init_out(float* __restrict__ out, int n) {
  int i = blockIdx.x * blockDim.x + threadIdx.x;
  if (i < n) out[i] = -INFINITY;
}

extern "C" void kernel_launch(void* const* d_in, const int* in_sizes, int n_in,
                              void* d_out, int out_size, void* d_ws, size_t ws_size,
                              hipStream_t stream) {
  (void)in_sizes; (void)n_in; (void)d_ws; (void)ws_size;
  const float* x = (const float*)d_in[0];   // [256, 2048]
  const float* M = (const float*)d_in[1];   // [2048, 2048]
  float* out = (float*)d_out;               // [256, 2048]

  init_out<<<dim3((out_size + 255) / 256), 256, 0, stream>>>(out, out_size);

  dim3 grid(N_DIM / I_TILE, B_DIM / B_TILE, K_SPLIT);  // 32 x 4 x 4 = 512 WGs
  tropical_mv<<<grid, 256, 0, stream>>>(x, M, out);
}